// NeighborSelector_30975304138951
// MI455X (gfx1250) — compile-verified
//
#include <hip/hip_runtime.h>
#include <cfloat>

#define BATCH 16
#define NCAND 8192
#define DIM   512
#define EPSV  1e-8f

typedef float v2f __attribute__((ext_vector_type(2)));
typedef float v8f __attribute__((ext_vector_type(8)));

// ---------------------------------------------------------------------------
// Kernel 0: per-batch query scalars  qq = sum q^2 ; qgq = sum q*g*q
// ---------------------------------------------------------------------------
__global__ __launch_bounds__(256)
void qscal_kernel(const float* __restrict__ query,
                  const float* __restrict__ metric,
                  float* __restrict__ qscal) {
    __shared__ float s1[256], s2[256];
    const int b = blockIdx.x, tid = threadIdx.x;
    const int i = tid * 2;
    float q0 = query[b * DIM + i],     q1 = query[b * DIM + i + 1];
    float g0 = metric[b * DIM + i],    g1 = metric[b * DIM + i + 1];
    s1[tid] = q0 * q0 + q1 * q1;
    s2[tid] = q0 * q0 * g0 + q1 * q1 * g1;
    __syncthreads();
    for (int off = 128; off > 0; off >>= 1) {
        if (tid < off) { s1[tid] += s1[tid + off]; s2[tid] += s2[tid + off]; }
        __syncthreads();
    }
    if (tid == 0) { qscal[2 * b] = s1[0]; qscal[2 * b + 1] = s2[0]; }
}

// ---------------------------------------------------------------------------
// Kernel 1: streaming geometric scores via V_WMMA_F32_16X16X4_F32.
//
// Each wave owns a 16-candidate tile. Per iteration it streams 8 dims via one
// global_load_b128 per lane and issues two WMMAs per reduction (K remapped:
// WMMA_a K0..3 -> dims {d0,d0+1,d0+4,d0+5}, WMMA_b -> {d0+2,d0+3,d0+6,d0+7};
// A and B use the same permutation so the dot products are unchanged).
// B-column masking is baked into per-lane LDS base pointers (zeros/ones
// planes) selected once outside the loop -> no per-iteration cndmask.
// C/D layout: VGPR r holds (M = r + 8*half, N = lane&15).
// ---------------------------------------------------------------------------
__global__ __launch_bounds__(256)
void geo_score_kernel(const float* __restrict__ query,
                      const float* __restrict__ cand,
                      const float* __restrict__ metric,
                      const float* __restrict__ transport,
                      const float* __restrict__ qscal,
                      float* __restrict__ dist_out,
                      float* __restrict__ scores_out) {
    __shared__ __align__(16) float sh_q[DIM];
    __shared__ __align__(16) float sh_g[DIM];
    __shared__ __align__(16) float sh_qw[DIM];
    __shared__ __align__(16) float sh_aqw[DIM];
    __shared__ __align__(16) float sh_at[DIM];
    __shared__ __align__(16) float sh_zero[DIM];
    __shared__ __align__(16) float sh_one[DIM];
    __shared__ float sh_red[8][16][6];

    const int b   = blockIdx.y;
    const int tid = threadIdx.x;

    for (int i = tid; i < DIM; i += 256) {
        float q = query[b * DIM + i];
        float g = metric[b * DIM + i];
        float t = transport[b * DIM + i];
        float w = q * g;
        sh_q[i] = q; sh_g[i] = g; sh_qw[i] = w;
        sh_aqw[i] = fabsf(w); sh_at[i] = fabsf(t);
        sh_zero[i] = 0.f; sh_one[i] = 1.f;
    }
    __syncthreads();

    const int wave = tid >> 5;
    const int lane = tid & 31;
    const int half = lane >> 4;
    const int nmod = lane & 15;           // A: row M ; B/D: column N
    const bool is_n0 = (nmod == 0);
    const bool is_n1 = (nmod == 1);

    // Lane-static B base pointers (masking by address, not data).
    const float* pB_qw = is_n0 ? sh_qw  : sh_zero;                      // col0: qw
    const float* pB_1g = is_n0 ? sh_one : (is_n1 ? sh_g : sh_zero);     // col0: 1, col1: g
    const float* pB_aw = is_n0 ? sh_aqw : sh_zero;                      // col0: |qw|
    const float* pB_at = is_n0 ? sh_at  : sh_zero;                      // col0: |t|

    const int n0 = (blockIdx.x * 8 + wave) * 16;
    const float* rowp = cand + ((size_t)b * NCAND + (size_t)(n0 + nmod)) * DIM;

    v8f acc_dot = {0,0,0,0,0,0,0,0};   // col0: sum qw*c
    v8f acc_cn  = {0,0,0,0,0,0,0,0};   // col0: sum c^2   col1: sum g*c^2
    v8f acc_abs = {0,0,0,0,0,0,0,0};   // col0: sum |qw|*|c|
    v8f acc_spn = {0,0,0,0,0,0,0,0};   // col0: sum |t|*|c-q|
    v8f acc_dsq = {0,0,0,0,0,0,0,0};   // col1: sum g*(c-q)^2

    for (int i = 0; i < DIM / 8; ++i) {
        const int d0 = i * 8 + 4 * half;
        const float4 cv = *(const float4*)(rowp + d0);       // global_load_b128
        const float4 qv = *(const float4*)(sh_q   + d0);     // ds_load_b128
        const float4 bq = *(const float4*)(pB_qw  + d0);
        const float4 bg = *(const float4*)(pB_1g  + d0);
        const float4 ba = *(const float4*)(pB_aw  + d0);
        const float4 bt = *(const float4*)(pB_at  + d0);

        const float e0 = cv.x - qv.x, e1 = cv.y - qv.y;
        const float e2 = cv.z - qv.z, e3 = cv.w - qv.w;

        // ---- WMMA_a : dims {d0, d0+1, d0+4, d0+5} (components .x,.y) ----
        {
            v2f A_c  = { cv.x, cv.y };
            v2f A_c2 = { cv.x * cv.x, cv.y * cv.y };
            v2f A_ac = { fabsf(cv.x), fabsf(cv.y) };
            v2f A_ad = { fabsf(e0), fabsf(e1) };
            v2f A_d2 = { e0 * e0, e1 * e1 };
            v2f B_qw = { bq.x, bq.y };
            v2f B_1g = { bg.x, bg.y };
            v2f B_aw = { ba.x, ba.y };
            v2f B_at = { bt.x, bt.y };
            acc_dot = __builtin_amdgcn_wmma_f32_16x16x4_f32(false, A_c,  false, B_qw, (short)0, acc_dot, false, false);
            acc_cn  = __builtin_amdgcn_wmma_f32_16x16x4_f32(false, A_c2, false, B_1g, (short)0, acc_cn,  false, false);
            acc_abs = __builtin_amdgcn_wmma_f32_16x16x4_f32(false, A_ac, false, B_aw, (short)0, acc_abs, false, false);
            acc_spn = __builtin_amdgcn_wmma_f32_16x16x4_f32(false, A_ad, false, B_at, (short)0, acc_spn, false, false);
            acc_dsq = __builtin_amdgcn_wmma_f32_16x16x4_f32(false, A_d2, false, B_1g, (short)0, acc_dsq, false, false);
        }
        // ---- WMMA_b : dims {d0+2, d0+3, d0+6, d0+7} (components .z,.w) ----
        {
            v2f A_c  = { cv.z, cv.w };
            v2f A_c2 = { cv.z * cv.z, cv.w * cv.w };
            v2f A_ac = { fabsf(cv.z), fabsf(cv.w) };
            v2f A_ad = { fabsf(e2), fabsf(e3) };
            v2f A_d2 = { e2 * e2, e3 * e3 };
            v2f B_qw = { bq.z, bq.w };
            v2f B_1g = { bg.z, bg.w };
            v2f B_aw = { ba.z, ba.w };
            v2f B_at = { bt.z, bt.w };
            acc_dot = __builtin_amdgcn_wmma_f32_16x16x4_f32(false, A_c,  false, B_qw, (short)0, acc_dot, false, false);
            acc_cn  = __builtin_amdgcn_wmma_f32_16x16x4_f32(false, A_c2, false, B_1g, (short)0, acc_cn,  false, false);
            acc_abs = __builtin_amdgcn_wmma_f32_16x16x4_f32(false, A_ac, false, B_aw, (short)0, acc_abs, false, false);
            acc_spn = __builtin_amdgcn_wmma_f32_16x16x4_f32(false, A_ad, false, B_at, (short)0, acc_spn, false, false);
            acc_dsq = __builtin_amdgcn_wmma_f32_16x16x4_f32(false, A_d2, false, B_1g, (short)0, acc_dsq, false, false);
        }
    }

    if (is_n0) {
        #pragma unroll
        for (int r = 0; r < 8; ++r) {
            const int m = r + 8 * half;
            sh_red[wave][m][0] = acc_dot[r];
            sh_red[wave][m][1] = acc_cn[r];
            sh_red[wave][m][2] = acc_abs[r];
            sh_red[wave][m][3] = acc_spn[r];
        }
    }
    if (is_n1) {
        #pragma unroll
        for (int r = 0; r < 8; ++r) {
            const int m = r + 8 * half;
            sh_red[wave][m][4] = acc_cn[r];
            sh_red[wave][m][5] = acc_dsq[r];
        }
    }
    __syncthreads();

    if (tid < 128) {
        const int wv = tid >> 4, mr = tid & 15;
        const int n  = (blockIdx.x * 8 + wv) * 16 + mr;
        const float dot  = sh_red[wv][mr][0];
        const float cc   = sh_red[wv][mr][1];
        const float asum = sh_red[wv][mr][2];
        const float spin = sh_red[wv][mr][3];
        const float cgc  = sh_red[wv][mr][4];
        const float dsq  = sh_red[wv][mr][5];
        const float qq   = qscal[2 * b];
        const float qgq  = qscal[2 * b + 1];

        const float dist_sq = fmaxf(dsq, EPSV);
        const float dist    = sqrtf(dist_sq);
        const float qn      = sqrtf(qq);
        const float cn      = sqrtf(fmaxf(cc, 0.f));
        const float ch0     = dot / (qn * fmaxf(cn, EPSV) + EPSV);
        const float prod    = qgq * cgc;
        const float wedge   = sqrtf(fmaxf(prod - dot * dot, 0.f) + EPSV);
        const float ch1     = wedge / fmaxf(sqrtf(fmaxf(prod, 0.f)), EPSV);
        const float ch2     = asum * (1.0f / (float)DIM);
        const float ch3     = spin / (dist + 0.001f);
        const float ch4     = -1.0f / (dist_sq + 0.001f);

        dist_out[(size_t)b * NCAND + n] = dist;
        float* sc = scores_out + ((size_t)b * NCAND + n) * 6;
        sc[0] = ch0; sc[1] = ch1; sc[2] = ch2; sc[3] = ch3; sc[4] = ch4; sc[5] = 0.f;
    }
}

// ---------------------------------------------------------------------------
// Kernel 2: deterministic iterative top-k (matches JAX tie-break: lower index)
// ---------------------------------------------------------------------------
__device__ void topk_list(float* base, int stride, int k, bool find_max, bool restore,
                          int* out_idx, float* sval, int* sidx,
                          float* keepv, int* keepi, int tid) {
    for (int kk = 0; kk < k; ++kk) {
        float bv = find_max ? -FLT_MAX : FLT_MAX;
        int   bi = 0x7FFFFFFF;
        for (int i = tid; i < NCAND; i += 256) {
            const float v = base[(size_t)i * stride];
            const bool better = find_max ? (v > bv || (v == bv && i < bi))
                                         : (v < bv || (v == bv && i < bi));
            if (better) { bv = v; bi = i; }
        }
        sval[tid] = bv; sidx[tid] = bi;
        __syncthreads();
        for (int off = 128; off > 0; off >>= 1) {
            if (tid < off) {
                const float ov = sval[tid + off]; const int oi = sidx[tid + off];
                const float cv = sval[tid];       const int ci = sidx[tid];
                const bool better = find_max ? (ov > cv || (ov == cv && oi < ci))
                                             : (ov < cv || (ov == cv && oi < ci));
                if (better) { sval[tid] = ov; sidx[tid] = oi; }
            }
            __syncthreads();
        }
        if (tid == 0) {
            const int ci = sidx[0];
            keepv[kk] = sval[0]; keepi[kk] = ci;
            out_idx[kk] = ci;
            base[(size_t)ci * stride] = find_max ? -FLT_MAX : FLT_MAX;
        }
        __syncthreads();
    }
    if (restore && tid < k) base[(size_t)keepi[tid] * stride] = keepv[tid];
    __syncthreads();
}

__global__ __launch_bounds__(256)
void topk_kernel(float* __restrict__ dist, float* __restrict__ scores,
                 int* __restrict__ selidx) {
    __shared__ float sval[256]; __shared__ int sidx[256];
    __shared__ float keepv[32]; __shared__ int keepi[32];
    const int b = blockIdx.x, tid = threadIdx.x;
    topk_list(dist + (size_t)b * NCAND, 1, 32, false, false,
              selidx + b * 64, sval, sidx, keepv, keepi, tid);
    topk_list(scores + (size_t)b * NCAND * 6 + 0, 6, 16, true, true,
              selidx + b * 64 + 32, sval, sidx, keepv, keepi, tid);
    topk_list(scores + (size_t)b * NCAND * 6 + 1, 6, 16, true, true,
              selidx + b * 64 + 48, sval, sidx, keepv, keepi, tid);
}

// ---------------------------------------------------------------------------
// Kernel 3: gather selected embeddings/scores/indices into d_out
// ---------------------------------------------------------------------------
__global__ __launch_bounds__(128)
void gather_kernel(const float* __restrict__ cand,
                   const float* __restrict__ scores,
                   const int* __restrict__ selidx,
                   float* __restrict__ out_emb,
                   float* __restrict__ out_sc,
                   float* __restrict__ out_idx) {
    const int blk = blockIdx.x;
    const int b = blk >> 6, s = blk & 63, tid = threadIdx.x;
    const int idx = selidx[b * 64 + s];
    const float4* src = (const float4*)(cand + ((size_t)b * NCAND + idx) * DIM);
    float4* dst = (float4*)(out_emb + ((size_t)b * 64 + s) * DIM);
    dst[tid] = src[tid];                       // 128 * 4 = 512 floats
    if (tid < 6) {
        const float v = (tid == 5) ? (float)s * (1.0f / 64.0f)
                                   : scores[((size_t)b * NCAND + idx) * 6 + tid];
        out_sc[((size_t)b * 64 + s) * 6 + tid] = v;
    }
    if (tid == 6) out_idx[b * 64 + s] = (float)idx;
}

// ---------------------------------------------------------------------------
extern "C" void kernel_launch(void* const* d_in, const int* in_sizes, int n_in,
                              void* d_out, int out_size, void* d_ws, size_t ws_size,
                              hipStream_t stream) {
    const float* query     = (const float*)d_in[0];
    const float* cand      = (const float*)d_in[1];
    const float* metric    = (const float*)d_in[2];
    const float* transport = (const float*)d_in[3];

    float* ws     = (float*)d_ws;
    float* qscal  = ws;                                    // 32 floats (pad 64)
    float* dist   = ws + 64;                               // B*N
    float* scores = dist + (size_t)BATCH * NCAND;          // B*N*6
    int*   selidx = (int*)(scores + (size_t)BATCH * NCAND * 6);  // B*64

    float* out     = (float*)d_out;
    float* out_emb = out;                                  // B*64*512
    float* out_sc  = out + (size_t)BATCH * 64 * DIM;       // B*64*6
    float* out_idx = out_sc + (size_t)BATCH * 64 * 6;      // B*64

    qscal_kernel<<<BATCH, 256, 0, stream>>>(query, metric, qscal);
    geo_score_kernel<<<dim3(NCAND / 128, BATCH), 256, 0, stream>>>(
        query, cand, metric, transport, qscal, dist, scores);
    topk_kernel<<<BATCH, 256, 0, stream>>>(dist, scores, selidx);
    gather_kernel<<<BATCH * 64, 128, 0, stream>>>(cand, scores, selidx,
                                                  out_emb, out_sc, out_idx);
}